// ApproxEMD_71004399337520
// MI455X (gfx1250) — compile-verified
//
#include <hip/hip_runtime.h>

typedef __attribute__((ext_vector_type(2))) float v2f;
typedef __attribute__((ext_vector_type(8))) float v8f;

#define EMD_N   2048
#define EMD_D   3
#define TILES   (EMD_N / 16)   // 128 16-wide tiles
#define TPB     1024           // 32 wave32 waves
#define NWAVES  (TPB / 32)
#define EMD_EPS 1e-9f

// -(4^i) for i = 7..-1, then 0.0 (matches reversed(range(-2,8)))
__constant__ float c_expf[10] = {
    -16384.0f, -4096.0f, -1024.0f, -256.0f, -64.0f,
    -16.0f, -4.0f, -1.0f, -0.25f, 0.0f};

__device__ __forceinline__ v8f wmma_dot16x16(v2f a, v2f b) {
    v8f c = {};
    // D = A(16x4, f32) * B(4x16, f32) + 0  -> 16x16 f32 dot-product tile
    return __builtin_amdgcn_wmma_f32_16x16x4_f32(
        /*neg_a=*/false, a, /*neg_b=*/false, b,
        /*c_mod=*/(short)0, c, /*reuse_a=*/false, /*reuse_b=*/false);
}

__global__ void __launch_bounds__(TPB, 1)
approx_emd_kernel(const float* __restrict__ preds,
                  const float* __restrict__ labels,
                  float* __restrict__ partial) {
    __shared__ float4 sP[EMD_N];     // preds:  x,y,z,|p|^2
    __shared__ float4 sL[EMD_N];     // labels: x,y,z,|l|^2
    __shared__ float  sCost[EMD_N];  // per-target remaining capacity
    __shared__ float  sCur[EMD_N];   // per-source remaining currency
    __shared__ float  sColA[EMD_N];  // sum_n E ; later f[m]
    __shared__ float  sColB[EMD_N];  // sum_n E*cur[n]
    __shared__ float  sRed[NWAVES];

    const int tid  = threadIdx.x;
    const int lane = tid & 31;
    const int wid  = tid >> 5;
    const int b    = blockIdx.x;

    // ---- stage batch slice + squared norms into LDS ----
    for (int i = tid; i < EMD_N; i += TPB) {
        const float* p = preds  + ((size_t)b * EMD_N + i) * EMD_D;
        const float* l = labels + ((size_t)b * EMD_N + i) * EMD_D;
        float px = p[0], py = p[1], pz = p[2];
        float lx = l[0], ly = l[1], lz = l[2];
        sP[i] = make_float4(px, py, pz, px * px + py * py + pz * pz);
        sL[i] = make_float4(lx, ly, lz, lx * lx + ly * ly + lz * lz);
        sCost[i] = 1.0f;
        sCur[i]  = 1.0f;
    }

    float outAcc = 0.0f;
    const int half8 = (lane >> 4) << 3;  // C/D rows: +0 (lanes 0-15) / +8 (16-31)
    const int l15   = lane & 15;
    const bool lo   = (lane < 16);

    for (int step = 0; step < 10; ++step) {
        const float ef = c_expf[step];
        __syncthreads();

        // ---- pass 1: column sums A[m], B[m] over all sources ----
        for (int jt = wid; jt < TILES; jt += NWAVES) {   // wave owns columns
            const int cbase = jt << 4;
            const float4 Lp = sL[cbase + l15];
            v2f bop;
            bop.x = lo ? Lp.x : Lp.z;
            bop.y = lo ? Lp.y : 0.0f;
            const float lm2 = Lp.w;
            float accA = 0.0f, accB = 0.0f;
            for (int rt = 0; rt < TILES; ++rt) {
                const int rbase = rt << 4;
                const float4 Pp = sP[rbase + l15];
                v2f aop;
                aop.x = lo ? Pp.x : Pp.z;
                aop.y = lo ? Pp.y : 0.0f;
                v8f dot = wmma_dot16x16(aop, bop);
                const int rh = rbase + half8;
                #pragma unroll
                for (int v = 0; v < 8; ++v) {
                    float dd = sP[rh + v].w + lm2 - 2.0f * dot[v];
                    float e  = __expf(ef * dd);
                    accA += e;
                    accB += e * sCur[rh + v];
                }
            }
            accA += __shfl_xor(accA, 16, 32);   // combine row halves
            accB += __shfl_xor(accB, 16, 32);
            if (lo) {
                sColA[cbase + lane] = accA;
                sColB[cbase + lane] = accB;
            }
        }
        __syncthreads();

        // ---- per-target auction scalars ----
        for (int m = tid; m < EMD_N; m += TPB) {
            float A = sColA[m], Bs = sColB[m], c = sCost[m];
            float inv1 = 1.0f / (c * A + EMD_EPS);
            float S2   = c * Bs * inv1;
            float wt   = fminf(c / (S2 + EMD_EPS), 1.0f);
            sColA[m]   = c * wt * inv1;              // f[m]
            sCost[m]   = fmaxf(c - S2 * wt, 0.0f);
        }
        __syncthreads();

        // ---- pass 2: row sums over all targets; output + currency ----
        for (int rt = wid; rt < TILES; rt += NWAVES) {  // wave owns rows
            const int rbase = rt << 4;
            const float4 Pp = sP[rbase + l15];
            v2f aop;
            aop.x = lo ? Pp.x : Pp.z;
            aop.y = lo ? Pp.y : 0.0f;
            float sEF[8], sEFD[8];
            #pragma unroll
            for (int v = 0; v < 8; ++v) { sEF[v] = 0.0f; sEFD[v] = 0.0f; }
            for (int jt = 0; jt < TILES; ++jt) {
                const int cbase = jt << 4;
                const float4 Lp = sL[cbase + l15];
                v2f bop;
                bop.x = lo ? Lp.x : Lp.z;
                bop.y = lo ? Lp.y : 0.0f;
                v8f dot = wmma_dot16x16(aop, bop);
                const float lm2 = Lp.w;
                const float fm  = sColA[cbase + l15];   // f[col]
                const int rh = rbase + half8;
                #pragma unroll
                for (int v = 0; v < 8; ++v) {
                    float dd = sP[rh + v].w + lm2 - 2.0f * dot[v];
                    float e  = __expf(ef * dd);
                    float t  = e * fm;
                    sEF[v]  += t;
                    sEFD[v] += t * dd;
                }
            }
            #pragma unroll
            for (int v = 0; v < 8; ++v) {
                float a = sEF[v], od = sEFD[v];
                #pragma unroll
                for (int off = 1; off < 16; off <<= 1) {  // 16-lane butterfly
                    a  += __shfl_xor(a,  off, 32);
                    od += __shfl_xor(od, off, 32);
                }
                if (l15 == 0) {   // lanes 0 and 16 own rows rbase+v, rbase+8+v
                    const int row = rbase + half8 + v;
                    float cu = sCur[row];
                    outAcc   += cu * od;                       // sum bids*d
                    sCur[row] = fmaxf(cu * (1.0f - a), 0.0f);  // currency update
                }
            }
        }
    }

    // ---- block reduction of outAcc; one partial per batch (no atomics) ----
    #pragma unroll
    for (int off = 16; off > 0; off >>= 1)
        outAcc += __shfl_xor(outAcc, off, 32);
    if (lane == 0) sRed[wid] = outAcc;
    __syncthreads();
    if (wid == 0) {
        float v = (lane < NWAVES) ? sRed[lane] : 0.0f;
        #pragma unroll
        for (int off = 16; off > 0; off >>= 1)
            v += __shfl_xor(v, off, 32);
        if (lane == 0) partial[b] = v;
    }
}

__global__ void finalize_kernel(const float* __restrict__ partial,
                                float* __restrict__ out, int B) {
    float s = 0.0f;
    for (int i = 0; i < B; ++i) s += partial[i];  // fixed order -> deterministic
    out[0] = s;
}

extern "C" void kernel_launch(void* const* d_in, const int* in_sizes, int n_in,
                              void* d_out, int out_size, void* d_ws, size_t ws_size,
                              hipStream_t stream) {
    const float* preds  = (const float*)d_in[0];
    const float* labels = (const float*)d_in[1];
    float* out = (float*)d_out;
    float* ws  = (float*)d_ws;
    const int B = in_sizes[0] / (EMD_N * EMD_D);

    approx_emd_kernel<<<dim3(B), dim3(TPB), 0, stream>>>(preds, labels, ws);
    finalize_kernel<<<dim3(1), dim3(1), 0, stream>>>(ws, out, B);
}